// RDFT_2018634629349
// MI455X (gfx1250) — compile-verified
//
#include <hip/hip_runtime.h>
#include <hip/hip_bf16.h>
#include <math.h>

// ---------------------------------------------------------------------------
// Problem constants (from reference)
// ---------------------------------------------------------------------------
#define BATCH   8
#define NTOK    16384
#define DIMC    192
#define RC      16
#define MTOK    64
#define LN_EPS  1e-5f
#define NRM_EPS 1e-12f

// main-kernel tiling
#define ROWS    128          // rows of x per block
#define NWAVES  8            // 8 waves * 16 rows = 128
#define THREADS 256

// padded LDS strides (bank-conflict-free WMMA fragment reads, 64 banks).
// All strides are multiples of 4 floats so 16-byte async B128 writes stay
// naturally aligned in LDS.
#define XS_STRIDE 196        // x tile rows       (196 % 64 == 4; 784B % 16 == 0)
#define KB_STRIDE 20         // k rows            (20L mod 64 is a permutation)
#define VS_STRIDE 200        // v rows (K dim)    (200 % 64 == 8)
#define QS_STRIDE 20         // q tile rows
#define AS_STRIDE 68         // attn tile rows    (68 % 64 == 4; scalar DS only)

typedef float v2f __attribute__((ext_vector_type(2)));
typedef float v8f __attribute__((ext_vector_type(8)));

// workspace layout (floats)
#define WS_K_OFF  0                          // kb:   B * M * RC   = 8192
#define WS_V_OFF  (BATCH * MTOK * RC)        // vref: B * M * DIM  = 98304

__device__ __forceinline__ float gelu_exact(float x) {
    return 0.5f * x * (1.0f + erff(x * 0.70710678118654752f));
}

// CDNA5 async global->LDS copy, 16 bytes per lane.  ASYNCcnt-tracked DMA:
// no VGPR round-trip, no LOADcnt/DScnt pressure.  LDS address = low 32 bits
// of the generic shared pointer.
__device__ __forceinline__ void async_copy_b128(float* lds_dst, const float* gsrc) {
    unsigned int l = (unsigned int)(uintptr_t)lds_dst;
    asm volatile("global_load_async_to_lds_b128 %0, %1, off"
                 :: "v"(l), "v"(gsrc) : "memory");
}
__device__ __forceinline__ void async_wait_all() {
    asm volatile("s_wait_asynccnt 0x0" ::: "memory");
}

// ---------------------------------------------------------------------------
// Kernel A: per-batch precompute of k = l2norm(LN(dict@wk)) and
// v = gelu(conv3tap(dict@wv + b)).  One block per batch; tiny work.
// ---------------------------------------------------------------------------
__global__ __launch_bounds__(THREADS)
void prep_kernel(const float* __restrict__ dict,
                 const float* __restrict__ wk_w, const float* __restrict__ wk_b,
                 const float* __restrict__ wv_w, const float* __restrict__ wv_b,
                 const float* __restrict__ kn_g, const float* __restrict__ kn_b,
                 const float* __restrict__ conv_w, const float* __restrict__ conv_b,
                 float* __restrict__ ws) {
    extern __shared__ float smem[];
    float* ds = smem;                    // 64*192 = 12288 dict tile
    float* vl = ds + MTOK * DIMC;        // 64*192 = 12288 raw v
    float* kl = vl + MTOK * DIMC;        // 64*16  = 1024  raw k

    const int b   = blockIdx.x;
    const int tid = threadIdx.x;
    const float* db = dict + (size_t)b * MTOK * DIMC;
    float* wsK = ws + WS_K_OFF + (size_t)b * MTOK * RC;
    float* wsV = ws + WS_V_OFF + (size_t)b * MTOK * DIMC;

    // stage dict[b] into LDS via async DMA (contiguous, 16B chunks)
    for (int i4 = tid; i4 < (MTOK * DIMC) / 4; i4 += THREADS)
        async_copy_b128(ds + i4 * 4, db + i4 * 4);
    async_wait_all();
    __syncthreads();

    // k_raw = dict @ wk + b   (64 x 16)
    for (int o = tid; o < MTOK * RC; o += THREADS) {
        const int m = o / RC, r = o % RC;
        float acc = wk_b[r];
        #pragma unroll 8
        for (int j = 0; j < DIMC; ++j) acc += ds[m * DIMC + j] * wk_w[j * RC + r];
        kl[m * RC + r] = acc;
    }
    __syncthreads();

    // per-row LN + l2norm on k, store to workspace
    if (tid < MTOK) {
        const int m = tid;
        float mu = 0.f;
        #pragma unroll
        for (int r = 0; r < RC; ++r) mu += kl[m * RC + r];
        mu *= (1.0f / RC);
        float var = 0.f;
        #pragma unroll
        for (int r = 0; r < RC; ++r) { float d = kl[m * RC + r] - mu; var += d * d; }
        var *= (1.0f / RC);
        const float rs = rsqrtf(var + LN_EPS);
        float tmp[RC];
        float nrm = 0.f;
        #pragma unroll
        for (int r = 0; r < RC; ++r) {
            float t = (kl[m * RC + r] - mu) * rs * kn_g[r] + kn_b[r];
            tmp[r] = t; nrm += t * t;
        }
        const float inv = 1.0f / fmaxf(sqrtf(nrm), NRM_EPS);
        #pragma unroll
        for (int r = 0; r < RC; ++r) wsK[m * RC + r] = tmp[r] * inv;
    }

    // v_raw = dict @ wv + b   (64 x 192)
    for (int o = tid; o < MTOK * DIMC; o += THREADS) {
        const int m = o / DIMC, c = o % DIMC;
        float acc = wv_b[c];
        #pragma unroll 8
        for (int j = 0; j < DIMC; ++j) acc += ds[m * DIMC + j] * wv_w[j * DIMC + c];
        vl[o] = acc;
    }
    __syncthreads();

    // depthwise 3-tap conv along M (kw=1 column of the 3x3 kernel) + GELU
    for (int o = tid; o < MTOK * DIMC; o += THREADS) {
        const int m = o / DIMC, c = o % DIMC;
        const float w0 = conv_w[c * 9 + 0 * 3 + 1];
        const float w1 = conv_w[c * 9 + 1 * 3 + 1];
        const float w2 = conv_w[c * 9 + 2 * 3 + 1];
        float acc = conv_b[c] + w1 * vl[m * DIMC + c];
        if (m > 0)        acc += w0 * vl[(m - 1) * DIMC + c];
        if (m < MTOK - 1) acc += w2 * vl[(m + 1) * DIMC + c];
        wsV[o] = gelu_exact(acc);
    }
}

// ---------------------------------------------------------------------------
// Kernel B: streaming attention kernel.  One block = 128 rows of x.
// All matmuls via v_wmma_f32_16x16x4_f32 (fp32 kept end-to-end; kernel is
// HBM-bound at ~18 flop/byte so f32 WMMA rate is not the limiter).  All LDS
// staging uses CDNA5 async global->LDS DMA.
// ---------------------------------------------------------------------------
__global__ __launch_bounds__(THREADS)
void attn_kernel(const float* __restrict__ x,
                 const float* __restrict__ wq_w, const float* __restrict__ wq_b,
                 const float* __restrict__ qn_g, const float* __restrict__ qn_b,
                 const float* __restrict__ temperature,
                 const float* __restrict__ ws,
                 float* __restrict__ out) {
    extern __shared__ float smem[];
    float* xs  = smem;                           // 128 * 196
    float* wqs = xs  + ROWS * XS_STRIDE;         // 192 * 16
    float* kbs = wqs + DIMC * RC;                // 64 * 20
    float* vs  = kbs + MTOK * KB_STRIDE;         // 64 * 200
    float* qs  = vs  + MTOK * VS_STRIDE;         // 8 * 16 * 20
    float* as_ = qs  + NWAVES * 16 * QS_STRIDE;  // 8 * 16 * 68

    const int tid  = threadIdx.x;
    const int wave = tid >> 5;                   // wave32
    const int lane = tid & 31;
    const int lm   = lane & 15;                  // lane % 16
    const int lh   = lane >> 4;                  // 0 / 1 half-wave

    const int tilesPerBatch = NTOK / ROWS;       // 128
    const int b       = blockIdx.x / tilesPerBatch;
    const int rowTile = blockIdx.x % tilesPerBatch;
    const int nBase   = rowTile * ROWS;

    const float invTemp = 1.0f / fmaxf(temperature[0], 0.5f);
    const float* wsK = ws + WS_K_OFF + (size_t)b * MTOK * RC;
    const float* wsV = ws + WS_V_OFF + (size_t)b * MTOK * DIMC;

    // ---- stage operands into LDS via async global->LDS DMA (16B/lane) ----
    for (int i4 = tid; i4 < (DIMC * RC) / 4; i4 += THREADS)
        async_copy_b128(wqs + i4 * 4, wq_w + i4 * 4);
    for (int i4 = tid; i4 < (MTOK * RC) / 4; i4 += THREADS) {
        const int i = i4 * 4;
        async_copy_b128(kbs + (i / RC) * KB_STRIDE + (i % RC), wsK + i);
    }
    for (int i4 = tid; i4 < (MTOK * DIMC) / 4; i4 += THREADS) {
        const int i = i4 * 4;
        async_copy_b128(vs + (i / DIMC) * VS_STRIDE + (i % DIMC), wsV + i);
    }
    {
        const float* xb = x + ((size_t)b * NTOK + nBase) * DIMC;
        for (int i4 = tid; i4 < (ROWS * DIMC) / 4; i4 += THREADS) {
            const int i = i4 * 4;
            async_copy_b128(xs + (i / DIMC) * XS_STRIDE + (i % DIMC), xb + i);
        }
    }
    async_wait_all();
    __syncthreads();

    const int rbase = wave * 16;
    float* qsw = qs  + wave * 16 * QS_STRIDE;
    float* asw = as_ + wave * 16 * AS_STRIDE;

    // ---- stage 2: q tile (16x16) = x_tile(16x192) @ wq(192x16) ----
    {
        v8f acc = {};
        #pragma unroll 4
        for (int k0 = 0; k0 < DIMC; k0 += 4) {
            v2f a, bf;
            const int ka = k0 + lh * 2;
            a.x  = xs[(rbase + lm) * XS_STRIDE + ka + 0];
            a.y  = xs[(rbase + lm) * XS_STRIDE + ka + 1];
            bf.x = wqs[(ka + 0) * RC + lm];
            bf.y = wqs[(ka + 1) * RC + lm];
            acc = __builtin_amdgcn_wmma_f32_16x16x4_f32(
                false, a, false, bf, (short)0, acc, false, false);
        }
        #pragma unroll
        for (int j = 0; j < 8; ++j)
            qsw[(j + 8 * lh) * QS_STRIDE + lm] = acc[j];
    }
    __syncthreads();

    // ---- stage 3: per-row bias + LayerNorm + l2norm on q ----
    if (lane < 16) {
        float* row = qsw + lane * QS_STRIDE;
        float v[RC];
        float mu = 0.f;
        #pragma unroll
        for (int c = 0; c < RC; ++c) { v[c] = row[c] + wq_b[c]; mu += v[c]; }
        mu *= (1.0f / RC);
        float var = 0.f;
        #pragma unroll
        for (int c = 0; c < RC; ++c) { float d = v[c] - mu; var += d * d; }
        var *= (1.0f / RC);
        const float rs = rsqrtf(var + LN_EPS);
        float nrm = 0.f;
        #pragma unroll
        for (int c = 0; c < RC; ++c) {
            v[c] = (v[c] - mu) * rs * qn_g[c] + qn_b[c];
            nrm += v[c] * v[c];
        }
        const float inv = 1.0f / fmaxf(sqrtf(nrm), NRM_EPS);
        #pragma unroll
        for (int c = 0; c < RC; ++c) row[c] = v[c] * inv;
    }
    __syncthreads();

    // ---- stage 4: attn logits = q(16x16) @ k^T(16x64), pre-scaled ----
    #pragma unroll
    for (int t = 0; t < MTOK / 16; ++t) {
        v8f acc = {};
        #pragma unroll
        for (int k0 = 0; k0 < RC; k0 += 4) {
            v2f a, bf;
            const int ka = k0 + lh * 2;
            a.x  = qsw[lm * QS_STRIDE + ka + 0];
            a.y  = qsw[lm * QS_STRIDE + ka + 1];
            bf.x = kbs[(t * 16 + lm) * KB_STRIDE + ka + 0];
            bf.y = kbs[(t * 16 + lm) * KB_STRIDE + ka + 1];
            acc = __builtin_amdgcn_wmma_f32_16x16x4_f32(
                false, a, false, bf, (short)0, acc, false, false);
        }
        #pragma unroll
        for (int j = 0; j < 8; ++j)
            asw[(j + 8 * lh) * AS_STRIDE + t * 16 + lm] = acc[j] * invTemp;
    }
    __syncthreads();

    // ---- stage 5: per-row softmax over M=64, in LDS ----
    if (lane < 16) {
        float* row = asw + lane * AS_STRIDE;
        float mx = -INFINITY;
        #pragma unroll 8
        for (int m = 0; m < MTOK; ++m) mx = fmaxf(mx, row[m]);
        float s = 0.f;
        #pragma unroll 8
        for (int m = 0; m < MTOK; ++m) { float e = __expf(row[m] - mx); row[m] = e; s += e; }
        const float inv = 1.0f / s;
        #pragma unroll 8
        for (int m = 0; m < MTOK; ++m) row[m] *= inv;
    }
    __syncthreads();

    // ---- cooperative coalesced writeback of attn (second output) ----
    {
        float* attnOut = out + (size_t)BATCH * NTOK * DIMC;   // outputs concatenated
        for (int f = tid; f < NWAVES * 16 * MTOK; f += THREADS) {
            const int w2 = f / (16 * MTOK);
            const int r  = (f / MTOK) & 15;
            const int m  = f % MTOK;
            const int n  = nBase + w2 * 16 + r;
            attnOut[((size_t)b * NTOK + n) * MTOK + m] =
                as_[(w2 * 16 + r) * AS_STRIDE + m];
        }
    }

    // ---- stage 6: out tile = attn(16x64) @ v(64x192), direct stores ----
    #pragma unroll 2
    for (int ct = 0; ct < DIMC / 16; ++ct) {
        v8f acc = {};
        #pragma unroll 4
        for (int k0 = 0; k0 < MTOK; k0 += 4) {
            v2f a, bf;
            const int ka = k0 + lh * 2;
            a.x  = asw[lm * AS_STRIDE + ka + 0];
            a.y  = asw[lm * AS_STRIDE + ka + 1];
            bf.x = vs[(ka + 0) * VS_STRIDE + ct * 16 + lm];
            bf.y = vs[(ka + 1) * VS_STRIDE + ct * 16 + lm];
            acc = __builtin_amdgcn_wmma_f32_16x16x4_f32(
                false, a, false, bf, (short)0, acc, false, false);
        }
        #pragma unroll
        for (int j = 0; j < 8; ++j) {
            const int n = nBase + wave * 16 + j + 8 * lh;
            out[((size_t)b * NTOK + n) * DIMC + ct * 16 + lm] = acc[j];
        }
    }
}

// ---------------------------------------------------------------------------
extern "C" void kernel_launch(void* const* d_in, const int* in_sizes, int n_in,
                              void* d_out, int out_size, void* d_ws, size_t ws_size,
                              hipStream_t stream) {
    const float* x      = (const float*)d_in[0];
    const float* dict   = (const float*)d_in[1];
    // d_in[2] = H, d_in[3] = W (unused)
    const float* wq_w   = (const float*)d_in[4];
    const float* wq_b   = (const float*)d_in[5];
    const float* wk_w   = (const float*)d_in[6];
    const float* wk_b   = (const float*)d_in[7];
    const float* wv_w   = (const float*)d_in[8];
    const float* wv_b   = (const float*)d_in[9];
    const float* qn_g   = (const float*)d_in[10];
    const float* qn_b   = (const float*)d_in[11];
    const float* kn_g   = (const float*)d_in[12];
    const float* kn_b   = (const float*)d_in[13];
    const float* conv_w = (const float*)d_in[14];
    const float* conv_b = (const float*)d_in[15];
    const float* temp   = (const float*)d_in[16];
    float* out = (float*)d_out;
    float* ws  = (float*)d_ws;

    const size_t prepShmem = (size_t)(MTOK * DIMC * 2 + MTOK * RC) * sizeof(float);
    const size_t mainShmem = (size_t)(ROWS * XS_STRIDE + DIMC * RC +
                                      MTOK * KB_STRIDE + MTOK * VS_STRIDE +
                                      NWAVES * 16 * QS_STRIDE +
                                      NWAVES * 16 * AS_STRIDE) * sizeof(float);

    (void)hipFuncSetAttribute((const void*)prep_kernel,
        hipFuncAttributeMaxDynamicSharedMemorySize, (int)prepShmem);
    (void)hipFuncSetAttribute((const void*)attn_kernel,
        hipFuncAttributeMaxDynamicSharedMemorySize, (int)mainShmem);

    prep_kernel<<<BATCH, THREADS, prepShmem, stream>>>(
        dict, wk_w, wk_b, wv_w, wv_b, kn_g, kn_b, conv_w, conv_b, ws);

    const int nBlocks = BATCH * (NTOK / ROWS);   // 1024
    attn_kernel<<<nBlocks, THREADS, mainShmem, stream>>>(
        x, wq_w, wq_b, qn_g, qn_b, temp, ws, out);
}